// Bnb8bitParametrization_85203561218574
// MI455X (gfx1250) — compile-verified
//
#include <hip/hip_runtime.h>
#include <hip/hip_bf16.h>

// int8_vectorwise_dequant: out[r][c] = q[r][c] * row_stats[r] * (1/127)
// Fixed shape 8192 x 8192, int32 in -> float32 out.
//
// Roofline: 64Mi elems x 8 B = 537 MB mandatory HBM traffic -> ~23 us floor
// at 23.3 TB/s; ~2 VALU ops/elem is >30x below the VALU roof. Pure streaming:
// 128-bit NT loads/stores, 4 independent b128 chains per thread, wave32.
//
// Layout facts exploited (all compile-time):
//  - 8192 cols / 4 = 2048 float4-vectors per row.
//  - Each block handles 1024 consecutive vectors (256 thr x 4 iters),
//    block base is 1024-aligned  =>  whole block sits inside ONE row,
//    row = blockIdx.x >> 1  => scale is block-uniform (pure scalar path).
//  - 16Mi vectors == 16384 blocks exactly => no bounds checks needed.

typedef int   i32x4 __attribute__((ext_vector_type(4)));
typedef float f32x4 __attribute__((ext_vector_type(4)));

#define DQ_BLOCK       256
#define DQ_UNROLL      4
#define DQ_VECS_BLOCK  (DQ_BLOCK * DQ_UNROLL)   // 1024 vectors per block

__global__ __launch_bounds__(DQ_BLOCK)
void Bnb8bitDequant_kernel(const i32x4* __restrict__ q,
                           const float* __restrict__ row_stats,
                           f32x4* __restrict__ out)
{
    const int base = blockIdx.x * DQ_VECS_BLOCK + threadIdx.x;

    // Block-uniform per-row scale: one s_load_b32 + s_mul_f32 per block.
    const float s = row_stats[blockIdx.x >> 1] * (1.0f / 127.0f);

    i32x4 d[DQ_UNROLL];
#pragma unroll
    for (int it = 0; it < DQ_UNROLL; ++it) {
        // 4 independent streaming b128 loads -> single s_clause, 4 in flight.
        d[it] = __builtin_nontemporal_load(&q[base + it * DQ_BLOCK]);
    }

#pragma unroll
    for (int it = 0; it < DQ_UNROLL; ++it) {
        f32x4 r;
        r.x = (float)d[it].x * s;
        r.y = (float)d[it].y * s;
        r.z = (float)d[it].z * s;
        r.w = (float)d[it].w * s;
        // Streaming write: written once, never re-read -> keep out of L2/WGP$.
        __builtin_nontemporal_store(r, &out[base + it * DQ_BLOCK]);
    }
}

extern "C" void kernel_launch(void* const* d_in, const int* in_sizes, int n_in,
                              void* d_out, int out_size, void* d_ws, size_t ws_size,
                              hipStream_t stream)
{
    const i32x4* q         = (const i32x4*)d_in[0]; // quantized_param, int32[8192*8192]
    const float* row_stats = (const float*)d_in[1]; // float32[8192]
    f32x4*       out       = (f32x4*)d_out;         // float32[8192*8192]

    // out_size = 64Mi elements; each block covers 4096 elements exactly.
    const int blocks = out_size / (4 * DQ_VECS_BLOCK);   // 16384

    Bnb8bitDequant_kernel<<<blocks, DQ_BLOCK, 0, stream>>>(q, row_stats, out);
}